// OptimizedRegressionLoss_22342419873868
// MI455X (gfx1250) — compile-verified
//
#include <hip/hip_runtime.h>
#include <hip/hip_bf16.h>
#include <math.h>

// ---------------------------------------------------------------------------
// Fused 3D SSIM + L1 loss for (4,1,128,128,128) f32 volumes, MI455X (gfx1250).
// One workgroup = one 16^3 output tile; 26^3 halo region staged in LDS.
// Separable 11-tap box sums run as band-matrix multiplies on
// V_WMMA_F32_16X16X4_F32 (exact f32 accumulation). All LDS operands use a
// K-minor, pad-to-28 layout so every B-fragment gather is one unconditional
// ds_load_b64 — no EXEC-masked loads, no per-WMMA dscnt stalls.
// ---------------------------------------------------------------------------

typedef __attribute__((ext_vector_type(2))) float v2f;
typedef __attribute__((ext_vector_type(8))) float v8f;

#define TILE     16
#define HALO     5
#define REG      26                  // TILE + 2*HALO
#define KP       28                  // K padded to multiple of 4
#define NROW_S   688                 // source (h,d) rows: 676 + 12 pad (W tail)
#define NROW_B1  (TILE * REG)        // 416 rows (w,d) for H pass
#define NROW_B2  (TILE * TILE)       // 256 rows (h,w) for D pass
#define SZ_S     (NROW_S * KP)       // 19264 floats per staged volume
#define SZ_B1    (NROW_B1 * KP)      // 11648
#define SZ_B2    (NROW_B2 * KP)      // 7168
#define NTHREADS 256
#define NBLOCKS  2048                // 4 batches * 8*8*8 tiles

// Band matrix entry: out index m sums region rows j = m..m+10 (11-tap box)
__device__ __forceinline__ float bandval(int m, int k) {
  return ((unsigned)(k - m) <= 10u) ? 1.0f : 0.0f;
}

// One 16x16 tile of Band(16x26) * Src(26xN): preload 7 B fragments, then a
// clean 7-deep WMMA accumulation chain.
template <typename SrcF>
__device__ __forceinline__ v8f band_tile(const v2f av[7], SrcF src) {
  v2f bv[7];
  #pragma unroll
  for (int kc = 0; kc < 7; ++kc) bv[kc] = src(kc);
  v8f c = {0.f, 0.f, 0.f, 0.f, 0.f, 0.f, 0.f, 0.f};
  #pragma unroll
  for (int kc = 0; kc < 7; ++kc)
    c = __builtin_amdgcn_wmma_f32_16x16x4_f32(false, av[kc], false, bv[kc],
                                              (short)0, c, false, false);
  return c;
}

extern "C" __global__ __launch_bounds__(NTHREADS)
void ssim3d_fused_kernel(const float* __restrict__ pred,
                         const float* __restrict__ targ,
                         float* __restrict__ partial) {
  extern __shared__ float smem[];
  float* sP   = smem;            // [(h*26+d)*28 + w], rows >=676 & w>=26 zero
  float* sT   = sP + SZ_S;
  float* buf1 = sT + SZ_S;       // [(w_out*26+d)*28 + h], h>=26 slots zero
  float* buf2 = buf1 + SZ_B1;    // [(h_out*16+w)*28 + d], d>=26 slots zero

  const int tid  = threadIdx.x;
  const int lrow = tid & 15;          // M row (lanes 0-15) / row (lanes 16-31)
  const int half = (tid >> 4) & 1;    // K sub-chunk selector per f32 layouts
  const int wave = tid >> 5;
  const int koff = half * 2;

  const int blk = blockIdx.x;
  const int b   = blk >> 9;           // batch
  const int rem = blk & 511;
  const int tz  = rem >> 6;           // D tile
  const int ty  = (rem >> 3) & 7;     // H tile
  const int tx  = rem & 7;            // W tile
  const int w0  = tx * TILE - HALO;
  const int h0  = ty * TILE - HALO;
  const int d0  = tz * TILE - HALO;
  const size_t base = (size_t)b * (128 * 128 * 128);

  // Band A fragments: depend only on lane; shared by all three passes.
  v2f av[7];
  #pragma unroll
  for (int kc = 0; kc < 7; ++kc) {
    const int k = kc * 4 + koff;
    av[kc].x = bandval(lrow, k);
    av[kc].y = bandval(lrow, k + 1);
  }

  // ---- Stage halo region (zero-padded: volume borders, k-pad, row-pad) ----
  for (int i = tid; i < SZ_S; i += NTHREADS) {
    const int row = i / KP;           // (h,d) flattened
    const int w   = i - row * KP;
    float pv = 0.0f, tv = 0.0f;
    if (row < REG * REG && w < REG) {
      const int h  = row / REG;
      const int d  = row - h * REG;
      const int gw = w0 + w, gh = h0 + h, gd = d0 + d;
      if (((unsigned)gw < 128u) & ((unsigned)gh < 128u) &
          ((unsigned)gd < 128u)) {
        const size_t g = base + (((size_t)gd * 128 + gh) * 128 + gw);
        pv = pred[g];
        tv = targ[g];
      }
    }
    sP[i] = pv;
    sT[i] = tv;
  }
  // Zero buf1+buf2 once: their k>=26 pad slots are never written afterwards.
  for (int i = tid; i < SZ_B1 + SZ_B2; i += NTHREADS) buf1[i] = 0.0f;

  float res[5][2][8];   // filtered sums: [quantity][tile slot][D row]

  #pragma unroll
  for (int q = 0; q < 5; ++q) {   // 0:p 1:t 2:p^2 3:t^2 4:p*t
    __syncthreads();
    // ---- W pass: Out rows = w_out(16), K = region W, N = (h,d) 676 ----
    for (int nt = wave; nt < 43; nt += 8) {
      const int n = nt * 16 + lrow;
      const float* pB = sP + n * KP + koff;
      const float* tB = sT + n * KP + koff;
      v8f c = band_tile(av, [&](int kc) -> v2f {
        const v2f p = *(const v2f*)(pB + kc * 4);
        const v2f t = *(const v2f*)(tB + kc * 4);
        return (q == 0) ? p
             : (q == 1) ? t
             : (q == 2) ? p * p
             : (q == 3) ? t * t
                        : p * t;
      });
      if (n < REG * REG) {
        const int h = n / REG;
        const int d = n - h * REG;
        float* o = buf1 + d * KP + h;
        #pragma unroll
        for (int r = 0; r < 8; ++r)
          o[(r + half * 8) * (REG * KP)] = c[r];
      }
    }
    __syncthreads();
    // ---- H pass: Out rows = h_out(16), K = region H, N = (w,d) 416 ----
    for (int nt = wave; nt < 26; nt += 8) {
      const int n = nt * 16 + lrow;
      const float* bB = buf1 + n * KP + koff;
      v8f c = band_tile(av, [&](int kc) -> v2f {
        return *(const v2f*)(bB + kc * 4);
      });
      const int w = n / REG;
      const int d = n - w * REG;
      float* o = buf2 + w * KP + d;
      #pragma unroll
      for (int r = 0; r < 8; ++r)
        o[(r + half * 8) * (TILE * KP)] = c[r];
    }
    __syncthreads();
    // ---- D pass: Out rows = d_out(16), K = region D, N = (h,w) 256 ----
    #pragma unroll
    for (int s = 0; s < 2; ++s) {
      const int nt = wave + 8 * s;      // == h_out row of this N tile
      const int n  = nt * 16 + lrow;
      const float* bB = buf2 + n * KP + koff;
      v8f c = band_tile(av, [&](int kc) -> v2f {
        return *(const v2f*)(bB + kc * 4);
      });
      #pragma unroll
      for (int r = 0; r < 8; ++r)
        res[q][s][r] = c[r];
    }
  }

  // ---- SSIM + L1 per owned voxel: (w=lrow, h=wave+8s, d=r+8*half) ----
  const float invK3 = 1.0f / 1331.0f;
  const float c1 = 1.0e-4f, c2 = 9.0e-4f;
  float acc_ssim = 0.0f, acc_l1 = 0.0f;
  #pragma unroll
  for (int s = 0; s < 2; ++s) {
    const int ho = wave + 8 * s;
    #pragma unroll
    for (int r = 0; r < 8; ++r) {
      const int dd = r + half * 8;
      const float mu_p = res[0][s][r] * invK3;
      const float mu_t = res[1][s][r] * invK3;
      const float ep2  = res[2][s][r] * invK3;
      const float et2  = res[3][s][r] * invK3;
      const float ept  = res[4][s][r] * invK3;
      const float mp2 = mu_p * mu_p, mt2 = mu_t * mu_t, mpt = mu_p * mu_t;
      const float sg_p = ep2 - mp2, sg_t = et2 - mt2, sg_pt = ept - mpt;
      const float num = (2.0f * mpt + c1) * (2.0f * sg_pt + c2);
      const float den = (mp2 + mt2 + c1) * (sg_p + sg_t + c2);
      acc_ssim += num / den;
      const int li = ((ho + HALO) * REG + (dd + HALO)) * KP + (lrow + HALO);
      acc_l1 += fabsf(sP[li] - sT[li]);
    }
  }

  // ---- Deterministic block reduction (reuse buf1) ----
  __syncthreads();
  float* red = buf1;
  red[tid]            = acc_ssim;
  red[NTHREADS + tid] = acc_l1;
  __syncthreads();
  for (int off = NTHREADS / 2; off > 0; off >>= 1) {
    if (tid < off) {
      red[tid]            += red[tid + off];
      red[NTHREADS + tid] += red[NTHREADS + tid + off];
    }
    __syncthreads();
  }
  if (tid == 0) {
    partial[2 * blk]     = red[0];         // ssim sum of this tile
    partial[2 * blk + 1] = red[NTHREADS];  // l1 sum of this tile
  }
}

extern "C" __global__ __launch_bounds__(NTHREADS)
void ssim3d_final_kernel(const float* __restrict__ partial,
                         float* __restrict__ out) {
  __shared__ float sm[2 * NTHREADS];
  const int tid = threadIdx.x;
  float ss = 0.0f, sl = 0.0f;
  for (int i = tid; i < NBLOCKS; i += NTHREADS) {
    ss += partial[2 * i];
    sl += partial[2 * i + 1];
  }
  sm[tid]            = ss;
  sm[NTHREADS + tid] = sl;
  __syncthreads();
  for (int off = NTHREADS / 2; off > 0; off >>= 1) {
    if (tid < off) {
      sm[tid]            += sm[tid + off];
      sm[NTHREADS + tid] += sm[NTHREADS + tid + off];
    }
    __syncthreads();
  }
  if (tid == 0) {
    const float invN = 1.0f / 8388608.0f;   // 4*128^3
    const float l1   = sm[NTHREADS] * invN;
    const float sls  = 1.0f - sm[0] * invN;
    out[0] = l1 + 0.5f * sls;   // total_loss
    out[1] = l1;                // l1_loss
    out[2] = sls;               // ssim_loss
    out[3] = 0.0f;              // reg_loss
  }
}

extern "C" void kernel_launch(void* const* d_in, const int* in_sizes, int n_in,
                              void* d_out, int out_size, void* d_ws,
                              size_t ws_size, hipStream_t stream) {
  (void)in_sizes; (void)n_in; (void)out_size; (void)ws_size;
  const float* pred = (const float*)d_in[0];
  const float* targ = (const float*)d_in[1];
  float* out     = (float*)d_out;
  float* partial = (float*)d_ws;  // 2048 * 2 floats = 16 KB scratch

  const size_t ldsBytes =
      (size_t)(2 * SZ_S + SZ_B1 + SZ_B2) * sizeof(float);   // 229376 B
  // Allow >64KB dynamic LDS (WGP has 320 KB); idempotent, capture-safe.
  hipFuncSetAttribute((const void*)ssim3d_fused_kernel,
                      hipFuncAttributeMaxDynamicSharedMemorySize,
                      (int)ldsBytes);

  ssim3d_fused_kernel<<<dim3(NBLOCKS), dim3(NTHREADS), ldsBytes, stream>>>(
      pred, targ, partial);
  ssim3d_final_kernel<<<dim3(1), dim3(NTHREADS), 0, stream>>>(partial, out);
}